// NNUE_67748814127512
// MI455X (gfx1250) — compile-verified
//
#include <hip/hip_runtime.h>

// ---------------------------------------------------------------------------
// NNUE head collapse for MI455X (gfx1250).
//
// out[b] = sum over valid (i,j) of RS[x_i*768 + x_j], where
//   RS[r] = sum_d tiles[r*64 + d]            (768*768 rows, 64 f32 each)
//   valid(v) : v<768 && v!=3 && (turn==1 || v!=443)
//
// Phase 1 (HBM-bound, 151 MB @ 23.3 TB/s ~ 6.5us): async-copy 16KB tile
// chunks into LDS (padded rows -> no DS bank conflicts), reduce 16 rows per
// wave with V_WMMA_F32_16X16X4_F32 against an all-ones B matrix.
// Phase 2: 1.3M gathers from the 2.25MB RS table (L2-resident), wave32
// butterfly reduction per batch.
// ---------------------------------------------------------------------------

typedef __attribute__((ext_vector_type(2))) float v2f;
typedef __attribute__((ext_vector_type(8))) float v8f;

// Signature of __builtin_amdgcn_global_load_async_to_lds_b128 (from hipcc
// diagnostic): param 0 is `int __vector(4) __device__ *` -> v4i in AS(1);
// LDS side is the AS(3) counterpart.
typedef int v4i_ __attribute__((vector_size(16)));
typedef __attribute__((address_space(1))) v4i_* gptr_v4i;
typedef __attribute__((address_space(3))) v4i_* lptr_v4i;

constexpr int kDIN   = 768;
constexpr int kF     = 36;
constexpr int kB     = 1024;
constexpr int kNPair = kF * kF;          // 1296
constexpr int kNRows = kDIN * kDIN;      // 589824 rows of 64 floats
constexpr int kRowsPerBlk   = 64;        // 64 rows * 256B = 16KB per block
constexpr int kLdsRowBytes  = 272;       // 256B + 16B pad -> bank-conflict free

#if defined(__has_builtin)
#if __has_builtin(__builtin_amdgcn_global_load_async_to_lds_b128) && \
    __has_builtin(__builtin_amdgcn_s_wait_asynccnt)
#define USE_ASYNC_LDS 1
#endif
#endif

// ---------------------------------------------------------------------------
// Phase 1: row sums of tiles via WMMA f32 16x16x4 with all-ones B.
// Block = 128 threads (4 waves); each wave reduces 16 rows.
// ---------------------------------------------------------------------------
__global__ __launch_bounds__(128) void rowsum_wmma(
    const float* __restrict__ tiles, float* __restrict__ rs) {
  __shared__ alignas(16) unsigned char smem[kRowsPerBlk * kLdsRowBytes];

  const int tid = threadIdx.x;
  const long long blockRow = (long long)blockIdx.x * kRowsPerBlk;
  const unsigned char* gbase =
      (const unsigned char*)tiles + blockRow * 256;  // 64 floats/row = 256B

  // Stage 16KB of tiles into LDS, inserting 16B pad every 256B row so that
  // the 16 A-fragment rows read by a wave land in distinct banks.
#pragma unroll
  for (int c = 0; c < 8; ++c) {
    const int goff = tid * 16 + c * 2048;            // 128 lanes * 16B * 8
    const int row  = goff >> 8;
    const int loff = row * kLdsRowBytes + (goff & 255);
#if USE_ASYNC_LDS
    __builtin_amdgcn_global_load_async_to_lds_b128(
        (gptr_v4i)(gbase + goff),
        (lptr_v4i)(smem + loff),
        /*offset=*/0, /*cpol=*/0);
#else
    *(float4*)(smem + loff) = *(const float4*)(gbase + goff);
#endif
  }
#if USE_ASYNC_LDS
  __builtin_amdgcn_s_wait_asynccnt(0);
#endif
  __syncthreads();

  const int wave  = tid >> 5;
  const int lane  = tid & 31;
  const int mrow  = lane & 15;            // A-matrix M index (16x4 layout)
  const int khalf = (lane >> 4) << 1;     // lanes 0-15: K={0,1}; 16-31: K={2,3}
  const unsigned char* lrow =
      smem + (wave * 16 + mrow) * kLdsRowBytes;

  v8f acc = {0.f, 0.f, 0.f, 0.f, 0.f, 0.f, 0.f, 0.f};
  v2f bones;
  bones.x = 1.0f;
  bones.y = 1.0f;

  // 16 WMMAs cover d = 0..63: D[m,n] += sum_k A[m,k] * 1  (every column n
  // holds the same row-sum, so extraction is trivial).
#pragma unroll
  for (int i = 0; i < 16; ++i) {
    const v2f a = *(const v2f*)(lrow + (i * 4 + khalf) * 4);  // 8B aligned
    acc = __builtin_amdgcn_wmma_f32_16x16x4_f32(
        /*neg_a=*/false, a, /*neg_b=*/false, bones,
        /*c_mod=*/(short)0, acc, /*reuse_a=*/false, /*reuse_b=*/false);
  }

  // C/D layout: VGPR j -> (M=j, N=lane) for lanes 0-15, (M=j+8, N=lane-16)
  // for lanes 16-31. Lane 0 holds rows 0..7, lane 16 holds rows 8..15.
  float* dst = rs + blockRow + wave * 16;
  if (lane == 0) {
    *(float4*)(dst + 0) = make_float4(acc[0], acc[1], acc[2], acc[3]);
    *(float4*)(dst + 4) = make_float4(acc[4], acc[5], acc[6], acc[7]);
  } else if (lane == 16) {
    *(float4*)(dst + 8)  = make_float4(acc[0], acc[1], acc[2], acc[3]);
    *(float4*)(dst + 12) = make_float4(acc[4], acc[5], acc[6], acc[7]);
  }
}

// ---------------------------------------------------------------------------
// Phase 2: per-batch pair gather-sum. One wave32 per batch, 8 batches/block.
// ---------------------------------------------------------------------------
__global__ __launch_bounds__(256) void gather_sum(
    const int* __restrict__ x, const int* __restrict__ turn,
    const float* __restrict__ rs, float* __restrict__ out) {
  __shared__ int sx[8][kF];

  const int wave = threadIdx.x >> 5;
  const int lane = threadIdx.x & 31;
  const int b = blockIdx.x * 8 + wave;

  const int* xrow = x + b * kF;
  const int t = turn[b];

  // Pre-mask: store -1 for invalid features (v>=768, v==3, and for black
  // turn also v==443, since the flipped zero-mask maps back to 443/3).
  for (int i = lane; i < kF; i += 32) {
    const int v = xrow[i];
    const bool ok = (v < kDIN) && (v != 3) && (t == 1 || v != 443);
    sx[wave][i] = ok ? v : -1;
  }
  __syncthreads();

  float sum = 0.0f;
  for (int p = lane; p < kNPair; p += 32) {
    const int i = p / kF;
    const int j = p - i * kF;
    const int vi = sx[wave][i];
    const int vj = sx[wave][j];
    if ((vi | vj) >= 0) sum += rs[vi * kDIN + vj];
  }

  // wave32 butterfly reduction
#pragma unroll
  for (int off = 16; off > 0; off >>= 1) sum += __shfl_down(sum, off, 32);

  if (lane == 0) out[b] = sum;
  if (blockIdx.x == 0 && threadIdx.x == 0) out[kB] = 0.0f;  // second output
}

// ---------------------------------------------------------------------------
extern "C" void kernel_launch(void* const* d_in, const int* in_sizes, int n_in,
                              void* d_out, int out_size, void* d_ws,
                              size_t ws_size, hipStream_t stream) {
  (void)in_sizes; (void)n_in; (void)out_size; (void)ws_size;
  const int*   x     = (const int*)d_in[0];    // (1024, 36) int32
  const int*   turn  = (const int*)d_in[1];    // (1024, 1)  int32
  const float* tiles = (const float*)d_in[2];  // 12*8*8*12*8*8*64 f32
  float* out = (float*)d_out;                  // 1025 f32
  float* rs  = (float*)d_ws;                   // 589824 f32 = 2.25 MB scratch

  rowsum_wmma<<<kNRows / kRowsPerBlk, 128, 0, stream>>>(tiles, rs);
  gather_sum<<<kB / 8, 256, 0, stream>>>(x, turn, rs, out);
}